// PhysicsLossCalculator_17454747091577
// MI455X (gfx1250) — compile-verified
//
#include <hip/hip_runtime.h>

// PhysicsLossCalculator on MI455X (gfx1250), compile-only loop.
// spectra_loss is statically 0 (all radial FFT bins empty) -> skip the FFT.
// Pass 1: per-column fused vorticity + vertical double-FD -> qg_pv + per-block
//         partial sums (variance terms, mass term). Pass 2: horizontal qg_pv
//         gradient partial sums. Pass 3: deterministic tree reduction (no
//         atomics). Wave reductions use V_WMMA_F32_16X16X4_F32.

#define LATN 721
#define LONN 1440
#define LVLS 13
#define BSZ  2
#define YX   (LATN * LONN)
#define NBX  ((YX + 255) / 256)          // 4056 blocks along y*x
#define NROWS 80                         // 26 sum + 26 sumsq + 2 mass + 26 grad

#define PI_F 3.14159265358979323846f
#define EARTH_R 6371000.0f
#define OMEGA_C 7.292e-05f

typedef __attribute__((ext_vector_type(2))) float v2f;
typedef __attribute__((ext_vector_type(8))) float v8f;

// Sum a float across all 32 lanes of the wave using the f32 WMMA.
// A(16x4): lane j<16 -> A[j][0..1]; lane j+16 -> A[j][2..3]. B = ones(4x16)
// => D[m][n] = sum_k A[m][k]. C/D layout: lanes 0-15 hold rows 0..7 in
// c[0..7], lanes 16-31 hold rows 8..15. c[0..7] summed + xor-16 partner's
// sum gives the exact f32 total over all 32 lanes.
__device__ __forceinline__ float wave_sum32(float val) {
#if __has_builtin(__builtin_amdgcn_wmma_f32_16x16x4_f32)
  v2f a; a[0] = val;  a[1] = 0.0f;
  v2f b; b[0] = 1.0f; b[1] = 1.0f;
  v8f c = {};
  c = __builtin_amdgcn_wmma_f32_16x16x4_f32(false, a, false, b,
                                            (short)0, c, false, false);
  float s = c[0] + c[1] + c[2] + c[3] + c[4] + c[5] + c[6] + c[7];
  s += __shfl_xor(s, 16, 32);
  return s;
#else
  for (int off = 16; off; off >>= 1) val += __shfl_xor(val, off, 32);
  return val;
#endif
}

// torch.gradient-style FD along lon (stride 1) at pointer p = &a[y][x].
__device__ __forceinline__ float fdx_lon(const float* p, int x, float inv_h) {
  if (x == 0)        return (p[1] - p[0]) * inv_h;
  if (x == LONN - 1) return (p[0] - p[-1]) * inv_h;
  return (p[1] - p[-1]) * (0.5f * inv_h);
}
// FD along lat (stride LONN).
__device__ __forceinline__ float fdy_lat(const float* p, int y, float inv_h) {
  if (y == 0)        return (p[LONN] - p[0]) * inv_h;
  if (y == LATN - 1) return (p[0] - p[-LONN]) * inv_h;
  return (p[LONN] - p[-LONN]) * (0.5f * inv_h);
}

// ---------------- init: dp, trapezoid weights, lat tables ----------------
__global__ void k_init(const float* __restrict__ plev, float* __restrict__ dpw,
                       float* __restrict__ ftab, float* __restrict__ invdx) {
  const int i = threadIdx.x;
  if (i < LATN) {
    float lat = -0.5f * PI_F + (float)i * (PI_F / (float)(LATN - 1));
    ftab[i] = 2.0f * OMEGA_C * sinf(lat);
    float cl = cosf(lat);
    if (cl < 1e-8f) cl = 1e-8f;
    invdx[i] = 1.0f / (EARTH_R * cl);
  }
  if (i == 0) {
    // dp = mean(diff(plev)) * 100 = (last - first)/(L-1) * 100
    dpw[0] = (plev[LVLS - 1] - plev[0]) / (float)(LVLS - 1) * 100.0f;
    float p[LVLS], w[LVLS];
    for (int l = 0; l < LVLS; l++) p[l] = plev[l] * 100.0f;
    w[0] = (p[1] - p[0]) * 0.5f;
    w[LVLS - 1] = (p[LVLS - 1] - p[LVLS - 2]) * 0.5f;
    for (int l = 1; l < LVLS - 1; l++) w[l] = (p[l + 1] - p[l - 1]) * 0.5f;
    float s = 0.0f;
    for (int l = 0; l < LVLS; l++) s += w[l];
    if (s < 1e-8f) s = 1e-8f;
    for (int l = 0; l < LVLS; l++) dpw[1 + l] = w[l] / s;
  }
}

// ---------------- pass 1: qg_pv + block partials -------------------------
// Partials matrix P[NROWS][NBX]:
//   rows  0..25 : sum(qg_pv)   per (b,l)
//   rows 26..51 : sum(qg_pv^2) per (b,l)
//   rows 52..53 : sum(col_div^2) per b
//   rows 54..79 : sum(|grad qg_pv|^2) per (b,l)   (written by pass 2)
__global__ void __launch_bounds__(256)
k_qgpv(const float* __restrict__ u, const float* __restrict__ v,
       const float* __restrict__ dpw, const float* __restrict__ ftab,
       const float* __restrict__ invdxt, float* __restrict__ qg,
       float* __restrict__ P) {
  const int b = blockIdx.y;                       // uniform per block
  const int t = blockIdx.x * blockDim.x + threadIdx.x;
  const bool active = t < YX;
  const int y = active ? t / LONN : 0;
  const int x = active ? t % LONN : 0;

  const float inv_dlat = (float)(LATN - 1) / PI_F;          // 1/dlat
  const float inv_dlon = (float)LONN / (2.0f * PI_F);       // 1/dlon
  const float invR = 1.0f / EARTH_R;
  const float inv_dx = invdxt[y];                           // 1/(R*cos)
  const float fcor = ftab[y];
  const float inv_dp = 1.0f / dpw[0];

  const size_t lstride = (size_t)YX;
  const size_t base = (size_t)b * LVLS * lstride + (size_t)y * LONN + x;

  float vort[LVLS];
  float cd = 0.0f;                                // weighted column divergence
#pragma unroll
  for (int l = 0; l < LVLS; l++) {
    const float* up = u + base + (size_t)l * lstride;
    const float* vp = v + base + (size_t)l * lstride;
    const float fdxv = fdx_lon(vp, x, inv_dlon);
    const float fdyu = fdy_lat(up, y, inv_dlat);
    const float fdxu = fdx_lon(up, x, inv_dlon);
    const float fdyv = fdy_lat(vp, y, inv_dlat);
    vort[l] = fdxv * inv_dx - fdyu * invR;
    cd += dpw[1 + l] * (fdxu * inv_dx + fdyv * invR);
  }

  // vertical double finite difference over pressure (all in registers)
  float vp1[LVLS], vpp[LVLS];
  vp1[0] = (vort[1] - vort[0]) * inv_dp;
  vp1[LVLS - 1] = (vort[LVLS - 1] - vort[LVLS - 2]) * inv_dp;
#pragma unroll
  for (int l = 1; l < LVLS - 1; l++)
    vp1[l] = (vort[l + 1] - vort[l - 1]) * (0.5f * inv_dp);
  vpp[0] = (vp1[1] - vp1[0]) * inv_dp;
  vpp[LVLS - 1] = (vp1[LVLS - 1] - vp1[LVLS - 2]) * inv_dp;
#pragma unroll
  for (int l = 1; l < LVLS - 1; l++)
    vpp[l] = (vp1[l + 1] - vp1[l - 1]) * (0.5f * inv_dp);

  // wave-level sums (all lanes get the result), then block-level LDS reduce
  float s1[LVLS], s2[LVLS];
#pragma unroll
  for (int l = 0; l < LVLS; l++) {
    const float q = vort[l] + fcor + 1e-4f * vpp[l];   // F0^2/N^2 = 1e-4
    if (active) qg[base + (size_t)l * lstride] = q;
    const float sv = active ? q : 0.0f;
    s1[l] = wave_sum32(sv);
    s2[l] = wave_sum32(sv * sv);
  }
  const float cdm = active ? cd : 0.0f;
  const float sm = wave_sum32(cdm * cdm);

  __shared__ float lds[27][8];
  const int wv = threadIdx.x >> 5;
  const int lane = threadIdx.x & 31;
  if (lane == 0) {
#pragma unroll
    for (int l = 0; l < LVLS; l++) {
      lds[2 * l][wv] = s1[l];
      lds[2 * l + 1][wv] = s2[l];
    }
    lds[26][wv] = sm;
  }
  __syncthreads();
  if (threadIdx.x < 27) {
    float tsum = 0.0f;
#pragma unroll
    for (int w = 0; w < 8; w++) tsum += lds[threadIdx.x][w];
    int row;
    if (threadIdx.x < 26) {
      const int l = threadIdx.x >> 1;
      row = (threadIdx.x & 1) ? (26 + b * LVLS + l) : (b * LVLS + l);
    } else {
      row = 52 + b;
    }
    P[(size_t)row * NBX + blockIdx.x] = tsum;
  }
}

// ---------------- pass 2: horizontal qg_pv gradient partials -------------
__global__ void __launch_bounds__(256)
k_grad(const float* __restrict__ qg, const float* __restrict__ invdxt,
       float* __restrict__ P) {
  const int bl = blockIdx.y;                      // (b,l) slice, uniform
  const int t = blockIdx.x * blockDim.x + threadIdx.x;
  const bool active = t < YX;
  const int y = active ? t / LONN : 0;
  const int x = active ? t % LONN : 0;

  const float inv_dlat = (float)(LATN - 1) / PI_F;
  const float inv_dlon = (float)LONN / (2.0f * PI_F);
  const float invR = 1.0f / EARTH_R;
  const float inv_dx = invdxt[y];

  const float* p = qg + (size_t)bl * YX + (size_t)y * LONN + x;
  const float gl = fdy_lat(p, y, inv_dlat) * invR;
  const float gx = fdx_lon(p, x, inv_dlon) * inv_dx;
  const float g = active ? (gl * gl + gx * gx) : 0.0f;

  const float s = wave_sum32(g);
  __shared__ float lds[8];
  if ((threadIdx.x & 31) == 0) lds[threadIdx.x >> 5] = s;
  __syncthreads();
  if (threadIdx.x == 0) {
    float tsum = 0.0f;
#pragma unroll
    for (int w = 0; w < 8; w++) tsum += lds[w];
    P[(size_t)(54 + bl) * NBX + blockIdx.x] = tsum;
  }
}

// ---------------- pass 3: reduce each partials row to one double ---------
__global__ void __launch_bounds__(256)
k_reduce(const float* __restrict__ P, double* __restrict__ acc) {
  const int r = blockIdx.x;                       // 0..NROWS-1
  const float* row = P + (size_t)r * NBX;
  float t = 0.0f;
  for (int i = threadIdx.x; i < NBX; i += 256) t += row[i];
  const float s = wave_sum32(t);
  __shared__ float lds[8];
  if ((threadIdx.x & 31) == 0) lds[threadIdx.x >> 5] = s;
  __syncthreads();
  if (threadIdx.x == 0) {
    float tot = 0.0f;
#pragma unroll
    for (int w = 0; w < 8; w++) tot += lds[w];
    acc[r] = (double)tot;
  }
}

// ---------------- finalize ----------------------------------------------
__global__ void k_final(const double* __restrict__ acc, float* __restrict__ out) {
  const double N = (double)YX;
  double vs = 0.0;
  for (int i = 0; i < 26; i++) {
    const double s = acc[i], ss = acc[26 + i];
    vs += (ss - s * s / N) / (N - 1.0);           // unbiased (ddof=1)
  }
  const double pv_var = vs / 26.0;
  const double mass = (acc[52] + acc[53]) / (2.0 * N);
  double gsum = 0.0;
  for (int i = 54; i < 80; i++) gsum += acc[i];
  const double grad_mean = gsum / (26.0 * N);
  out[0] = (float)(pv_var + 0.1 * grad_mean + mass);  // spectra_loss == 0
}

extern "C" void kernel_launch(void* const* d_in, const int* in_sizes, int n_in,
                              void* d_out, int out_size, void* d_ws, size_t ws_size,
                              hipStream_t stream) {
  const float* u = (const float*)d_in[0];
  const float* v = (const float*)d_in[1];
  const float* plev = (const float*)d_in[2];

  // workspace layout (every consumed byte is written every launch)
  double* acc  = (double*)d_ws;                        // 80 doubles
  float* dpw   = (float*)((char*)d_ws + 1024);         // dp + 13 weights
  float* ftab  = (float*)((char*)d_ws + 1152);         // Coriolis f(lat) [721]
  float* invdx = (float*)((char*)d_ws + 4096);         // 1/(R*cos(lat)) [721]
  float* P     = (float*)((char*)d_ws + 8192);         // partials [80][4056]
  float* qg    = (float*)((char*)d_ws + 8192 + (size_t)NROWS * NBX * 4 + 4096);

  k_init<<<1, 1024, 0, stream>>>(plev, dpw, ftab, invdx);

  dim3 g1(NBX, BSZ);
  k_qgpv<<<g1, 256, 0, stream>>>(u, v, dpw, ftab, invdx, qg, P);

  dim3 g2(NBX, BSZ * LVLS);
  k_grad<<<g2, 256, 0, stream>>>(qg, invdx, P);

  k_reduce<<<NROWS, 256, 0, stream>>>(P, acc);
  k_final<<<1, 1, 0, stream>>>(acc, (float*)d_out);
}